// Net_33045478376051
// MI455X (gfx1250) — compile-verified
//
#include <hip/hip_runtime.h>
#include <hip/hip_bf16.h>
#include <math.h>

#define N_NODES 100000
#define N_EDGES 3200000
#define N_ETOT  (N_EDGES + N_NODES)   // edges + self loops
#define F_IN    512
#define HID     16
#define N_CLS   16

typedef __attribute__((ext_vector_type(2))) float v2f;
typedef __attribute__((ext_vector_type(8))) float v8f;

// -------------------------------------------------------------------------
// Kernel 1: h = relu(x @ W1^T + b1) via V_WMMA_F32_16X16X4_F32.
// One wave -> one 16-row x 16-col tile of h. K loop over 512 in steps of 4.
// W1 (16x512 f32 = 32 KB) staged in LDS per block.
// -------------------------------------------------------------------------
__global__ __launch_bounds__(256) void gemm1_wmma_kernel(
    const float* __restrict__ x, const float* __restrict__ w1,
    const float* __restrict__ b1, float* __restrict__ h, int n_nodes)
{
    __shared__ float w_lds[HID * F_IN];   // 32 KB
    const int tid = threadIdx.x;

    // cooperative weight load: 8192 floats = 2048 float4, 256 threads -> 8 each
    {
        const float4* w4 = (const float4*)w1;
        float4* l4 = (float4*)w_lds;
        #pragma unroll
        for (int i = 0; i < 8; ++i) l4[tid + 256 * i] = w4[tid + 256 * i];
    }
    __syncthreads();

    const int wave = tid >> 5;                 // 0..7
    const int lane = tid & 31;
    const int tile = blockIdx.x * 8 + wave;    // 16-row tile index
    const int row0 = tile * 16;
    if (row0 >= n_nodes) return;

    const int m     = lane & 15;               // A: row-in-tile, B: column (hid)
    const int khalf = (lane >> 4) * 2;         // 0 for lanes 0-15, 2 for 16-31

    const float* xrow = x     + (size_t)(row0 + m) * F_IN;
    const float* wrow = w_lds + (size_t)m * F_IN;

    v8f c = {};
    #pragma unroll 4
    for (int k0 = 0; k0 < F_IN; k0 += 4) {
        v2f a, b;
        a.x = xrow[k0 + khalf];
        a.y = xrow[k0 + khalf + 1];
        b.x = wrow[k0 + khalf];
        b.y = wrow[k0 + khalf + 1];
        c = __builtin_amdgcn_wmma_f32_16x16x4_f32(
                /*neg_a=*/false, a, /*neg_b=*/false, b,
                /*c_mod=*/(short)0, c, /*reuse_a=*/false, /*reuse_b=*/false);
    }

    // C/D layout: VGPR r -> (M = r, lanes 0-15) and (M = r+8, lanes 16-31), N = lane&15
    const int rowadd = (lane >> 4) * 8;
    const float bias = b1[m];
    #pragma unroll
    for (int r = 0; r < 8; ++r) {
        const int   rr = row0 + r + rowadd;
        const float v  = c[r] + bias;
        h[(size_t)rr * HID + m] = v > 0.0f ? v : 0.0f;
    }
}

// -------------------------------------------------------------------------
// Row-normalize: hn[i] = h[i] / max(||h[i]||, 1e-12). One thread per node.
// -------------------------------------------------------------------------
__global__ __launch_bounds__(256) void normalize_kernel(
    const float* __restrict__ h, float* __restrict__ hn, int n)
{
    const int i = blockIdx.x * blockDim.x + threadIdx.x;
    if (i >= n) return;
    const float4* hp = (const float4*)(h + (size_t)i * HID);
    float4 v0 = hp[0], v1 = hp[1], v2 = hp[2], v3 = hp[3];
    float s = v0.x*v0.x + v0.y*v0.y + v0.z*v0.z + v0.w*v0.w
            + v1.x*v1.x + v1.y*v1.y + v1.z*v1.z + v1.w*v1.w
            + v2.x*v2.x + v2.y*v2.y + v2.z*v2.z + v2.w*v2.w
            + v3.x*v3.x + v3.y*v3.y + v3.z*v3.z + v3.w*v3.w;
    const float inv = 1.0f / fmaxf(sqrtf(s), 1e-12f);
    float4* op = (float4*)(hn + (size_t)i * HID);
    v0.x *= inv; v0.y *= inv; v0.z *= inv; v0.w *= inv;
    v1.x *= inv; v1.y *= inv; v1.z *= inv; v1.w *= inv;
    v2.x *= inv; v2.y *= inv; v2.z *= inv; v2.w *= inv;
    v3.x *= inv; v3.y *= inv; v3.z *= inv; v3.w *= inv;
    op[0] = v0; op[1] = v1; op[2] = v2; op[3] = v3;
}

// -------------------------------------------------------------------------
// Edge pass 1: exp_e[e] = exp(beta * dot(hn[src], hn[dst])); a_sum[dst] += exp_e.
// Half-wave (16 lanes) per edge; shfl_xor tree reduction of the 16-wide dot.
// No segment-max needed: |alpha| <= |beta|, softmax is shift invariant.
// Edges e >= n_edges are the self loops (src = dst = e - n_edges).
// -------------------------------------------------------------------------
__global__ __launch_bounds__(256) void edge_alpha_kernel(
    const float* __restrict__ hn,
    const long long* __restrict__ src, const long long* __restrict__ dst,
    const float* __restrict__ beta_ptr,
    float* __restrict__ exp_e, float* __restrict__ a_sum,
    int n_edges, int n_total)
{
    const int gid = blockIdx.x * blockDim.x + threadIdx.x;
    const int e = gid >> 4;
    const int f = gid & 15;
    if (e >= n_total) return;

    long long s, d;
    if (e < n_edges) { s = src[e]; d = dst[e]; }
    else             { s = d = (long long)(e - n_edges); }

    float a = hn[(size_t)s * HID + f] * hn[(size_t)d * HID + f];
    a += __shfl_xor(a, 1, 32);
    a += __shfl_xor(a, 2, 32);
    a += __shfl_xor(a, 4, 32);
    a += __shfl_xor(a, 8, 32);

    if (f == 0) {
        const float beta = beta_ptr ? beta_ptr[0] : 1.0f;
        const float ee = __expf(beta * a);
        exp_e[e] = ee;
        atomicAdd(a_sum + d, ee);
    }
}

// -------------------------------------------------------------------------
// Edge pass 2: acc[dst] += h[src] * (exp_e[e] / a_sum[dst]).
// Half-wave per edge, one feature per lane.
// -------------------------------------------------------------------------
__global__ __launch_bounds__(256) void edge_aggr_kernel(
    const float* __restrict__ h,
    const long long* __restrict__ src, const long long* __restrict__ dst,
    const float* __restrict__ exp_e, const float* __restrict__ a_sum,
    float* __restrict__ acc, int n_edges, int n_total)
{
    const int gid = blockIdx.x * blockDim.x + threadIdx.x;
    const int e = gid >> 4;
    const int f = gid & 15;
    if (e >= n_total) return;

    long long s, d;
    if (e < n_edges) { s = src[e]; d = dst[e]; }
    else             { s = d = (long long)(e - n_edges); }

    const float w = exp_e[e] / a_sum[d];   // a_sum > 0: self-loop always contributes
    atomicAdd(&acc[(size_t)d * HID + f], h[(size_t)s * HID + f] * w);
}

// -------------------------------------------------------------------------
// Head: logits = h @ W2^T + b2 ; out = log_softmax(logits). One thread/node.
// -------------------------------------------------------------------------
__global__ __launch_bounds__(256) void head_kernel(
    const float* __restrict__ h, const float* __restrict__ w2,
    const float* __restrict__ b2, float* __restrict__ out, int n)
{
    const int i = blockIdx.x * blockDim.x + threadIdx.x;
    if (i >= n) return;

    float hv[HID];
    {
        const float4* hp = (const float4*)(h + (size_t)i * HID);
        #pragma unroll
        for (int q = 0; q < 4; ++q) {
            float4 v = hp[q];
            hv[q*4+0] = v.x; hv[q*4+1] = v.y; hv[q*4+2] = v.z; hv[q*4+3] = v.w;
        }
    }

    float logits[N_CLS];
    float mx = -1e30f;
    #pragma unroll
    for (int c = 0; c < N_CLS; ++c) {
        float acc = b2[c];
        #pragma unroll
        for (int k = 0; k < HID; ++k) acc += w2[c * HID + k] * hv[k];
        logits[c] = acc;
        mx = fmaxf(mx, acc);
    }
    float sum = 0.0f;
    #pragma unroll
    for (int c = 0; c < N_CLS; ++c) sum += __expf(logits[c] - mx);
    const float lse = mx + __logf(sum);
    #pragma unroll
    for (int c = 0; c < N_CLS; ++c) out[(size_t)i * N_CLS + c] = logits[c] - lse;
}

// -------------------------------------------------------------------------
extern "C" void kernel_launch(void* const* d_in, const int* in_sizes, int n_in,
                              void* d_out, int out_size, void* d_ws, size_t ws_size,
                              hipStream_t stream)
{
    (void)in_sizes; (void)n_in; (void)out_size; (void)ws_size;

    const float*     x    = (const float*)d_in[0];
    const long long* ei   = (const long long*)d_in[1];   // int64 [2, E]
    const float*     w1   = (const float*)d_in[2];
    const float*     b1   = (const float*)d_in[3];
    const float*     w2   = (const float*)d_in[4];
    const float*     b2   = (const float*)d_in[5];
    const float*     beta2= (const float*)d_in[6];
    float*           out  = (float*)d_out;

    const long long* src = ei;
    const long long* dst = ei + (size_t)N_EDGES;

    // workspace layout (floats)
    float* h0    = (float*)d_ws;                 // N*16
    float* hn    = h0    + (size_t)N_NODES * HID;
    float* acc   = hn    + (size_t)N_NODES * HID;
    float* a_sum = acc   + (size_t)N_NODES * HID;
    float* exp_e = a_sum + (size_t)N_NODES;      // N_ETOT

    const int nodeBlocks = (N_NODES + 255) / 256;
    const int tileBlocks = ((N_NODES / 16) + 7) / 8;           // 782
    const int edgeBlocks = ((size_t)N_ETOT * 16 + 255) / 256;  // 206250

    // 1) h0 = relu(x @ W1^T + b1)  (WMMA)
    gemm1_wmma_kernel<<<tileBlocks, 256, 0, stream>>>(x, w1, b1, h0, N_NODES);

    // ---- prop 1 (beta = 1.0) ----
    normalize_kernel<<<nodeBlocks, 256, 0, stream>>>(h0, hn, N_NODES);
    hipMemsetAsync(a_sum, 0, (size_t)N_NODES * sizeof(float), stream);
    hipMemsetAsync(acc,   0, (size_t)N_NODES * HID * sizeof(float), stream);
    edge_alpha_kernel<<<edgeBlocks, 256, 0, stream>>>(hn, src, dst, (const float*)nullptr,
                                                      exp_e, a_sum, N_EDGES, N_ETOT);
    edge_aggr_kernel<<<edgeBlocks, 256, 0, stream>>>(h0, src, dst, exp_e, a_sum,
                                                     acc, N_EDGES, N_ETOT);

    // ---- prop 2 (beta = beta2[0]) ----  acc -> h0
    normalize_kernel<<<nodeBlocks, 256, 0, stream>>>(acc, hn, N_NODES);
    hipMemsetAsync(a_sum, 0, (size_t)N_NODES * sizeof(float), stream);
    hipMemsetAsync(h0,    0, (size_t)N_NODES * HID * sizeof(float), stream);
    edge_alpha_kernel<<<edgeBlocks, 256, 0, stream>>>(hn, src, dst, beta2,
                                                      exp_e, a_sum, N_EDGES, N_ETOT);
    edge_aggr_kernel<<<edgeBlocks, 256, 0, stream>>>(acc, src, dst, exp_e, a_sum,
                                                     h0, N_EDGES, N_ETOT);

    // 3) head: logits + log_softmax
    head_kernel<<<nodeBlocks, 256, 0, stream>>>(h0, w2, b2, out, N_NODES);
}